// CrossAttentionBlock3D_43215960933008
// MI455X (gfx1250) — compile-verified
//
#include <hip/hip_runtime.h>
#include <math.h>

#define C_DIM   512
#define S_DIM   4096
#define HEADS   8
#define HDIM    64
#define EPS     1e-5f

typedef __attribute__((ext_vector_type(2))) float v2f;
typedef __attribute__((ext_vector_type(8))) float v8f;
typedef __attribute__((ext_vector_type(4))) int   v4i;

__device__ __forceinline__ v8f wmma4(v2f a, v2f b, v8f c) {
  // D(16x16,f32) = A(16x4,f32) * B(4x16,f32) + C
  return __builtin_amdgcn_wmma_f32_16x16x4_f32(false, a, false, b, (short)0, c,
                                               false, false);
}

// Async global->LDS staging (gfx1250), with sync fallback if builtin absent.
#if defined(__AMDGCN__) && __has_builtin(__builtin_amdgcn_global_load_async_to_lds_b128)
#define USE_ASYNC_LDS 1
typedef __attribute__((address_space(1))) v4i gv4i;   // global (device) AS
typedef __attribute__((address_space(3))) v4i lv4i;   // LDS AS
__device__ __forceinline__ void async_cp16(const float* g, float* l) {
  __builtin_amdgcn_global_load_async_to_lds_b128((gv4i*)g, (lv4i*)l, 0, 0);
}
__device__ __forceinline__ void wait_async0() {
#if __has_builtin(__builtin_amdgcn_s_wait_asynccnt)
  __builtin_amdgcn_s_wait_asynccnt(0);
#else
  asm volatile("s_wait_asynccnt 0x0" ::: "memory");
#endif
}
#else
#define USE_ASYNC_LDS 0
__device__ __forceinline__ void async_cp16(const float* g, float* l) {
  *(float4*)l = *(const float4*)g;
}
__device__ __forceinline__ void wait_async0() {}
#endif

// ---------------------------------------------------------------------------
// GroupNorm statistics: 16 blocks = {x,ctx} x 8 groups (group channels are
// contiguous: 64ch * 4096 spatial floats).
// ---------------------------------------------------------------------------
__global__ void gn_stats(const float* __restrict__ x, const float* __restrict__ ctx,
                         float* __restrict__ stats) {
  int b = blockIdx.x;                 // 0..15
  int tensor = b >> 3, g = b & 7;
  const float* src = (tensor ? ctx : x) + (size_t)g * 64 * S_DIM;
  const int N = 64 * S_DIM;
  float s = 0.f, s2 = 0.f;
  for (int i = threadIdx.x; i < N; i += 256) {
    float v = src[i];
    s += v; s2 += v * v;
  }
  __shared__ float rs[256], rs2[256];
  rs[threadIdx.x] = s; rs2[threadIdx.x] = s2;
  __syncthreads();
  for (int off = 128; off > 0; off >>= 1) {
    if (threadIdx.x < off) {
      rs[threadIdx.x]  += rs[threadIdx.x + off];
      rs2[threadIdx.x] += rs2[threadIdx.x + off];
    }
    __syncthreads();
  }
  if (threadIdx.x == 0) {
    float mu  = rs[0] / N;
    float var = rs2[0] / N - mu * mu;
    stats[tensor * 16 + g]     = mu;
    stats[tensor * 16 + 8 + g] = rsqrtf(var + EPS);
  }
}

// ---------------------------------------------------------------------------
// Apply GroupNorm elementwise: dst = (src - mu)*rstd*w[c] + b[c]
// ---------------------------------------------------------------------------
__global__ void gn_apply(const float* __restrict__ src, const float* __restrict__ w,
                         const float* __restrict__ bia, const float* __restrict__ stats,
                         float* __restrict__ dst) {
  int idx = blockIdx.x * 256 + threadIdx.x;   // grid covers exactly C*S
  int c = idx >> 12;
  int g = c >> 6;
  float mu = stats[g], rstd = stats[8 + g];
  dst[idx] = (src[idx] - mu) * rstd * w[c] + bia[c];
}

// ---------------------------------------------------------------------------
// WMMA GEMM: out[M, S_DIM] = W[M, 512] * X[512, S_DIM] + bias
// Block = 256 threads (8 waves), output 128 rows x 64 cols per block.
// K is processed in 128-chunks staged transposed in LDS (xs[s][k]).
// Each wave: one 16-row strip x four 16-col tiles -> each A register pair
// feeds 4 WMMAs (4x less W traffic than 1 tile/wave).
// Store modes:
//   0 (NAT):  out0[m*S + s] = acc + bias[m] (+ resid)
//   1 (HT):   out0[(head*S + s)*64 + ch]   ([head][S][64] layout for q/k)
//   2 (KV):   rows <512 -> HT into out0 (k); rows >=512 -> NAT into out1 (v)
// ---------------------------------------------------------------------------
#define MODE_NAT 0
#define MODE_HT  1
#define MODE_KV  2
#define GK   128            // K chunk
#define GXS  130            // padded LDS row stride (even -> 8B-aligned pairs)

__global__ __launch_bounds__(256)
void gemm_wmma(const float* __restrict__ W, const float* __restrict__ X,
               const float* __restrict__ bias, float* __restrict__ out0,
               float* __restrict__ out1, const float* __restrict__ resid,
               int mode) {
  __shared__ float xs[64 * GXS];                // 33.3 KB
  const int s0 = blockIdx.x * 64;
  const int tid = threadIdx.x;
  const int wave = tid >> 5, lane = tid & 31;
  const int ln = lane & 15, hh = lane >> 4;
  const int m0 = (blockIdx.y * 8 + wave) * 16;
  const float* wrow = W + (size_t)(m0 + ln) * C_DIM;

  v8f acc[4] = {{}, {}, {}, {}};
  for (int kb = 0; kb < C_DIM; kb += GK) {
    __syncthreads();                            // previous chunk fully consumed
    {
      int sq = (tid & 15) * 4;                  // s: 0..60 step 4
      int kk = tid >> 4;                        // k: 0..15 per pass
      for (int kp = 0; kp < GK; kp += 16) {
        const float* gp = X + (size_t)(kb + kp + kk) * S_DIM + s0 + sq;
        float4 vv = *(const float4*)gp;
        xs[(sq + 0) * GXS + kp + kk] = vv.x;
        xs[(sq + 1) * GXS + kp + kk] = vv.y;
        xs[(sq + 2) * GXS + kp + kk] = vv.z;
        xs[(sq + 3) * GXS + kp + kk] = vv.w;
      }
    }
    __syncthreads();
    const float* wk = wrow + kb;
    #pragma unroll 4
    for (int t = 0; t < GK / 4; ++t) {
      int ko = 4 * t + 2 * hh;
      v2f a  = *(const v2f*)(wk + ko);
      v2f b0 = *(const v2f*)(xs + (ln +  0) * GXS + ko);
      v2f b1 = *(const v2f*)(xs + (ln + 16) * GXS + ko);
      v2f b2 = *(const v2f*)(xs + (ln + 32) * GXS + ko);
      v2f b3 = *(const v2f*)(xs + (ln + 48) * GXS + ko);
      acc[0] = wmma4(a, b0, acc[0]);
      acc[1] = wmma4(a, b1, acc[1]);
      acc[2] = wmma4(a, b2, acc[2]);
      acc[3] = wmma4(a, b3, acc[3]);
    }
  }

  #pragma unroll
  for (int j = 0; j < 4; ++j) {
    int s = s0 + j * 16 + ln;                   // D col
    #pragma unroll
    for (int r = 0; r < 8; ++r) {
      int m = m0 + r + 8 * hh;                  // D row
      float v = acc[j][r] + bias[m];
      if (mode == MODE_NAT) {
        if (resid) v += resid[(size_t)m * S_DIM + s];
        out0[(size_t)m * S_DIM + s] = v;
      } else if (mode == MODE_HT) {
        out0[((size_t)(m >> 6) * S_DIM + s) * HDIM + (m & 63)] = v;
      } else {                                  // MODE_KV
        if (m < C_DIM) out0[((size_t)(m >> 6) * S_DIM + s) * HDIM + (m & 63)] = v;
        else           out1[(size_t)(m - C_DIM) * S_DIM + s] = v;
      }
    }
  }
}

// ---------------------------------------------------------------------------
// Flash attention: block = 256 threads (8 waves) = 128 queries of one head.
// K/V 16-key tiles are double-buffered in LDS and prefetched with
// GLOBAL_LOAD_ASYNC_TO_LDS_B128 (ASYNCcnt) one tile ahead of the WMMA
// compute. Each wave: 16 QK^T WMMAs + online softmax + LDS transpose of P +
// 16 PV WMMAs per key tile.
// Q/K layout [head][S][64]; V layout [head*64][S] (natural kv-GEMM output ==
// the transposed form the PV B-operand needs).
// ---------------------------------------------------------------------------
__global__ __launch_bounds__(256)
void attn_kernel(const float* __restrict__ qb, const float* __restrict__ kb,
                 const float* __restrict__ vb, float* __restrict__ ob) {
  __shared__ float ktile[2][16 * 64];           // [key][ch], 4KB each
  __shared__ float vtile[2][64 * 16];           // [ch][key], 4KB each
  __shared__ float pl[8 * 256];                 // per-wave P transpose buffers

  const int hd = blockIdx.y;
  const int tid = threadIdx.x;
  const int wave = tid >> 5, lane = tid & 31;
  const int ln = lane & 15, hh = lane >> 4;
  const int q0 = (blockIdx.x * 8 + wave) * 16;
  float* plw = pl + wave * 256;

  const float* kbase = kb + (size_t)hd * S_DIM * HDIM;
  const float* vbase = vb + (size_t)hd * HDIM * S_DIM;

  // per-thread staging addresses (16B per buffer per thread)
  const int vrow = tid >> 2, vcol = (tid & 3) * 4;

  // Q tile resident in registers in A-operand layout
  const float* qrow = qb + ((size_t)hd * S_DIM + q0 + ln) * HDIM;
  v2f qr[16];
  #pragma unroll
  for (int t = 0; t < 16; ++t) qr[t] = *(const v2f*)(qrow + 4 * t + 2 * hh);

  v8f o0 = {}, o1 = {}, o2 = {}, o3 = {};
  float run_max[8], run_sum[8];
  #pragma unroll
  for (int r = 0; r < 8; ++r) { run_max[r] = -INFINITY; run_sum[r] = 0.f; }
  const float scale = 0.125f;                   // 1/sqrt(64)

  // prefetch tile 0
  async_cp16(kbase + 0 * HDIM + tid * 4, &ktile[0][tid * 4]);
  async_cp16(vbase + (size_t)vrow * S_DIM + 0 + vcol, &vtile[0][vrow * 16 + vcol]);

  for (int kt = 0; kt < S_DIM / 16; ++kt) {
    const int par = kt & 1;
    const float* kd = ktile[par];
    const float* vd = vtile[par];

    wait_async0();                              // this wave's stage done
    __syncthreads();                            // all waves' stages done;
                                                // also: par^1 readers finished
    if (kt + 1 < S_DIM / 16) {                  // prefetch next tile
      const int key1 = (kt + 1) * 16;
      async_cp16(kbase + (size_t)key1 * HDIM + tid * 4, &ktile[par ^ 1][tid * 4]);
      async_cp16(vbase + (size_t)vrow * S_DIM + key1 + vcol,
                 &vtile[par ^ 1][vrow * 16 + vcol]);
    }

    // S = Q * K^T (16 wmma steps over 64 channels), K B-operand from LDS
    v8f sacc = {};
    #pragma unroll
    for (int t = 0; t < 16; ++t) {
      v2f kr = *(const v2f*)(kd + ln * 64 + 4 * t + 2 * hh);
      sacc = wmma4(qr[t], kr, sacc);
    }

    // online softmax per query row (row m = r + 8*hh, keys across 16 lanes)
    float alpha[8];
    #pragma unroll
    for (int r = 0; r < 8; ++r) {
      float sv = sacc[r] * scale;
      float m = sv;
      m = fmaxf(m, __shfl_xor(m, 1));
      m = fmaxf(m, __shfl_xor(m, 2));
      m = fmaxf(m, __shfl_xor(m, 4));
      m = fmaxf(m, __shfl_xor(m, 8));
      float nm = fmaxf(run_max[r], m);
      alpha[r] = __expf(run_max[r] - nm);
      run_max[r] = nm;
      float p = __expf(sv - nm);
      float ps = p;
      ps += __shfl_xor(ps, 1);
      ps += __shfl_xor(ps, 2);
      ps += __shfl_xor(ps, 4);
      ps += __shfl_xor(ps, 8);
      run_sum[r] = run_sum[r] * alpha[r] + ps;
      sacc[r] = p;                              // keep P in sacc
    }
    #pragma unroll
    for (int r = 0; r < 8; ++r) {
      o0[r] = o0[r] * alpha[r];
      o1[r] = o1[r] * alpha[r];
      o2[r] = o2[r] * alpha[r];
      o3[r] = o3[r] * alpha[r];
    }

    // transpose P (D layout -> A layout) through this wave's LDS region;
    // same-wave LDS ops are in order, no barrier needed.
    #pragma unroll
    for (int r = 0; r < 8; ++r) plw[(r + 8 * hh) * 16 + ln] = sacc[r];
    v2f pa[4];
    #pragma unroll
    for (int t = 0; t < 4; ++t)
      pa[t] = *(const v2f*)(plw + ln * 16 + 4 * t + 2 * hh);

    // O += P * V (4 k-steps x 4 channel groups), V B-operand from LDS
    #pragma unroll
    for (int t = 0; t < 4; ++t) {
      const int ko = 4 * t + 2 * hh;
      v2f va = *(const v2f*)(vd + ( 0 + ln) * 16 + ko);
      v2f vc = *(const v2f*)(vd + (16 + ln) * 16 + ko);
      v2f ve = *(const v2f*)(vd + (32 + ln) * 16 + ko);
      v2f vf = *(const v2f*)(vd + (48 + ln) * 16 + ko);
      o0 = wmma4(pa[t], va, o0);
      o1 = wmma4(pa[t], vc, o1);
      o2 = wmma4(pa[t], ve, o2);
      o3 = wmma4(pa[t], vf, o3);
    }
    __syncthreads();                            // done reading kd/vd
  }

  // normalize + store to [C][S] for the proj GEMM
  #pragma unroll
  for (int r = 0; r < 8; ++r) {
    float inv = 1.f / run_sum[r];
    int s = q0 + r + 8 * hh;                    // query position (D row)
    ob[((size_t)hd * 64 +  0 + ln) * S_DIM + s] = o0[r] * inv;
    ob[((size_t)hd * 64 + 16 + ln) * S_DIM + s] = o1[r] * inv;
    ob[((size_t)hd * 64 + 32 + ln) * S_DIM + s] = o2[r] * inv;
    ob[((size_t)hd * 64 + 48 + ln) * S_DIM + s] = o3[r] * inv;
  }
}

// ---------------------------------------------------------------------------
extern "C" void kernel_launch(void* const* d_in, const int* in_sizes, int n_in,
                              void* d_out, int out_size, void* d_ws, size_t ws_size,
                              hipStream_t stream) {
  const float* x     = (const float*)d_in[0];
  const float* ctx   = (const float*)d_in[1];
  const float* nw    = (const float*)d_in[2];
  const float* nb    = (const float*)d_in[3];
  const float* ncw   = (const float*)d_in[4];
  const float* ncb   = (const float*)d_in[5];
  const float* qw    = (const float*)d_in[6];
  const float* qbias = (const float*)d_in[7];
  const float* kvw   = (const float*)d_in[8];
  const float* kvb   = (const float*)d_in[9];
  const float* pw    = (const float*)d_in[10];
  const float* pb    = (const float*)d_in[11];
  float* out = (float*)d_out;
  float* ws  = (float*)d_ws;

  const size_t T = (size_t)C_DIM * S_DIM;       // 2M floats
  float* h     = ws;            // normalized x           [C][S]
  float* cn    = ws + 1 * T;    // normalized context     [C][S]
  float* qws   = ws + 2 * T;    // q                      [head][S][64]
  float* kws   = ws + 3 * T;    // k                      [head][S][64]
  float* vws   = ws + 4 * T;    // v                      [head*64][S]
  float* aws   = ws + 5 * T;    // attention out          [C][S]
  float* stats = ws + 6 * T;    // 32 floats

  gn_stats<<<16, 256, 0, stream>>>(x, ctx, stats);
  gn_apply<<<T / 256, 256, 0, stream>>>(x,   nw,  nb,  stats,      h);
  gn_apply<<<T / 256, 256, 0, stream>>>(ctx, ncw, ncb, stats + 16, cn);

  dim3 gq(S_DIM / 64, C_DIM / 128);
  gemm_wmma<<<gq, 256, 0, stream>>>(qw, h, qbias, qws, nullptr, nullptr, MODE_HT);
  dim3 gkv(S_DIM / 64, (2 * C_DIM) / 128);
  gemm_wmma<<<gkv, 256, 0, stream>>>(kvw, cn, kvb, kws, vws, nullptr, MODE_KV);

  dim3 ga(S_DIM / 128, HEADS);
  attn_kernel<<<ga, 256, 0, stream>>>(qws, kws, vws, aws);

  gemm_wmma<<<gq, 256, 0, stream>>>(pw, aws, pb, out, nullptr, x, MODE_NAT);
}